// MetricBiasUpdater_5111011082726
// MI455X (gfx1250) — compile-verified
//
#include <hip/hip_runtime.h>
#include <hip/hip_bf16.h>

// ---------------------------------------------------------------------------
// MetricBiasUpdater for MI455X (gfx1250, wave32)
//
// Roofline: output/input stream of B_prev/B_next = 134 MB -> ~5.8us @ 23.3TB/s.
// Total FLOPs ~1.7 GF => purely HBM bound. G (1MB) + G_sq (32KB) are kept in
// workspace and stay L2-resident (192MB L2), so the Gram WMMA math rides for
// free under the B_prev stream. FP32 WMMA (V_WMMA_F32_16X16X4_F32) is chosen
// because the reference is f32 and we are not compute-limited.
// ---------------------------------------------------------------------------

#define D_MODEL   1024
#define GEOM_DIM  32
#define NB        4
#define NN        2048
#define CLAMP_V   10.0f

typedef __attribute__((ext_vector_type(2))) float v2f;
typedef __attribute__((ext_vector_type(8))) float v8f;

// ---------------------------------------------------------------------------
// Kernel 1: G[row,k] = dot(H[row,:], W[k,:]);  G_sq[row] = sum_k G[row,k]^2
// One wave per row (32 lanes == GEOM_DIM). H row elements broadcast across
// lanes (same address -> one fetch); W (128KB) is L2/WGP$-resident.
// ---------------------------------------------------------------------------
__global__ __launch_bounds__(256)
void geom_proj_kernel(const float* __restrict__ H,
                      const float* __restrict__ W,
                      float* __restrict__ G,
                      float* __restrict__ Gsq)
{
    const int wave = (blockIdx.x * blockDim.x + threadIdx.x) >> 5;   // row id
    const int lane = threadIdx.x & 31;                               // k id
    const float* __restrict__ hrow = H + (size_t)wave * D_MODEL;
    const float* __restrict__ wrow = W + (size_t)lane * D_MODEL;

    float acc = 0.0f;
    #pragma unroll 4
    for (int d = 0; d < D_MODEL; d += 4) {
        float4 h = *reinterpret_cast<const float4*>(hrow + d);
        float4 w = *reinterpret_cast<const float4*>(wrow + d);
        acc = fmaf(h.x, w.x, acc);
        acc = fmaf(h.y, w.y, acc);
        acc = fmaf(h.z, w.z, acc);
        acc = fmaf(h.w, w.w, acc);
    }
    G[(size_t)wave * GEOM_DIM + lane] = acc;

    // wave32 butterfly reduction of acc^2 -> G_sq[row]
    float sq = acc * acc;
    #pragma unroll
    for (int m = 16; m > 0; m >>= 1)
        sq += __shfl_xor(sq, m, 32);
    if (lane == 0) Gsq[wave] = sq;
}

// ---------------------------------------------------------------------------
// Kernel 2: per-wave 16x16 tile.
//   gram = G_i(16x32) @ G_j^T(32x16) via 8x V_WMMA_F32_16X16X4_F32
//   dist = relu(Gsq_i + Gsq_j - 2*gram)
//   out  = clamp(alpha*B_prev - beta*dist, +-10)
//
// 32-bit A(16x4) fragment layout (ISA 7.12.2): lanes 0-15 carry M=lane,
// VGPR0/1 = K=kb+0/kb+1; lanes 16-31 carry M=lane-16, VGPR0/1 = K=kb+2/kb+3.
// B(4x16) mirrors with N in place of M. C/D: VGPR r, lane l<16 -> (M=r, N=l),
// lane l>=16 -> (M=r+8, N=l-16).
// ---------------------------------------------------------------------------
__global__ __launch_bounds__(256)
void bias_update_kernel(const float* __restrict__ G,
                        const float* __restrict__ Gsq,
                        const float* __restrict__ Bprev,
                        float* __restrict__ Bnext,
                        const float* __restrict__ alpha_p,
                        const float* __restrict__ beta_p)
{
    const int lane = threadIdx.x & 31;
    const int wave = (blockIdx.x * blockDim.x + threadIdx.x) >> 5; // 0..65535
    const int b    = wave >> 14;         // / (128*128)
    const int rem  = wave & 16383;
    const int ti   = rem >> 7;           // row-tile  0..127
    const int tj   = rem & 127;          // col-tile  0..127

    const float* __restrict__ Gb  = G    + (size_t)b * NN * GEOM_DIM;
    const float* __restrict__ Qb  = Gsq  + (size_t)b * NN;
    const float* __restrict__ Bp  = Bprev + (size_t)b * NN * NN;
    float*       __restrict__ Bo  = Bnext + (size_t)b * NN * NN;

    const int half = lane >> 4;          // 0: K pair {0,1}; 1: K pair {2,3}
    const int l    = lane & 15;
    const int rowA = ti * 16 + l;        // A-matrix row (i)
    const int rowB = tj * 16 + l;        // B-matrix column (j) = row of G

    v8f c = {};
    #pragma unroll
    for (int kb = 0; kb < GEOM_DIM; kb += 4) {
        const int kk = kb + 2 * half;
        v2f a  = *reinterpret_cast<const v2f*>(Gb + (size_t)rowA * GEOM_DIM + kk);
        v2f bm = *reinterpret_cast<const v2f*>(Gb + (size_t)rowB * GEOM_DIM + kk);
        // 8 args: (neg_a, A, neg_b, B, c_mod, C, reuse_a, reuse_b)
        c = __builtin_amdgcn_wmma_f32_16x16x4_f32(
                /*neg_a=*/false, a, /*neg_b=*/false, bm,
                /*c_mod=*/(short)0, c, /*reuse_a=*/false, /*reuse_b=*/false);
    }

    const float alpha = *alpha_p;
    const float beta  = *beta_p;

    const int   j     = tj * 16 + l;     // output column for this lane
    const float gsq_j = Qb[j];           // column term (constant over VGPRs)
    const int   i0    = ti * 16 + 8 * half;

    #pragma unroll
    for (int r = 0; r < 8; ++r) {
        const int i = i0 + r;            // output row for C VGPR r
        float dist = Qb[i] + gsq_j - 2.0f * c[r];
        dist = fmaxf(dist, 0.0f);
        float out = fmaf(-beta, dist, alpha * Bp[(size_t)i * NN + j]);
        out = fminf(fmaxf(out, -CLAMP_V), CLAMP_V);
        Bo[(size_t)i * NN + j] = out;
    }
}

// ---------------------------------------------------------------------------
// Launch: inputs in setup_inputs() order: H, B_prev, W, alpha, beta
// ---------------------------------------------------------------------------
extern "C" void kernel_launch(void* const* d_in, const int* in_sizes, int n_in,
                              void* d_out, int out_size, void* d_ws, size_t ws_size,
                              hipStream_t stream)
{
    const float* H     = (const float*)d_in[0];
    const float* Bprev = (const float*)d_in[1];
    const float* W     = (const float*)d_in[2];
    const float* alpha = (const float*)d_in[3];
    const float* beta  = (const float*)d_in[4];
    float*       Bnext = (float*)d_out;

    // Workspace: G [4*2048*32] floats, then Gsq [4*2048] floats (~1.06 MB)
    float* G   = (float*)d_ws;
    float* Gsq = G + (size_t)NB * NN * GEOM_DIM;

    // Kernel 1: 8192 rows, 1 wave per row, 8 waves (256 thr) per block
    {
        const int rows   = NB * NN;              // 8192
        const int blocks = rows / 8;             // 1024
        geom_proj_kernel<<<blocks, 256, 0, stream>>>(H, W, G, Gsq);
    }

    // Kernel 2: 4 * 128 * 128 = 65536 tiles, 1 wave per tile, 8 waves/block
    {
        const int tiles  = NB * (NN / 16) * (NN / 16); // 65536
        const int blocks = tiles / 8;                  // 8192
        bias_update_kernel<<<blocks, 256, 0, stream>>>(G, Gsq, Bprev, Bnext,
                                                       alpha, beta);
    }
}